// ArccosHessianCalculator_46213848105705
// MI455X (gfx1250) — compile-verified
//
#include <hip/hip_runtime.h>
#include <hip/hip_bf16.h>

typedef float v2f __attribute__((ext_vector_type(2)));
typedef float v8f __attribute__((ext_vector_type(8)));

#define DIM 128

// Per-m specialized tile loop. M is a compile-time constant so all
// coefficient/operand selection is resolved at compile time; the only
// per-lane conditionals left are cndmask-class selects (no EXEC changes,
// as required around WMMA).
//
// Each wave owns output tile-row i0 = wave*16 and walks the 8 j-tiles in
// rotated order jt = (wave + t) & 7, so the diagonal tile is always t==0:
// only that WMMA takes the precomputed diagonal C; the other 7 take C=0.
template <int M>
__device__ __forceinline__ void hess_tiles(
    const float* __restrict__ sh1, const float* __restrict__ sh2,
    float rn1, float rn2, float cc,
    float* __restrict__ base, int wave, int mi, bool hi)
{
    //  M=0 (H11*inv11): s=rn1^2,   A=[s*u1, s*u2, -3c*s*u1], B=[u2,u1,u1], diag= c*s
    //  M=1 (H12*inv12): s=rn1*rn2, A=[s*u1, s*u2,  -c*s*u2], B=[u1,u2,u1], diag=-s
    //  M=2 (H22*inv22): s=rn2^2,   A=[s*u1, s*u2, -3c*s*u2], B=[u2,u1,u2], diag= c*s
    const float s    = (M == 0) ? rn1 * rn1 : ((M == 1) ? rn1 * rn2 : rn2 * rn2);
    const float diag = (M == 1) ? -s : cc * s;

    const int   i0  = wave << 4;
    const float u1i = sh1[i0 + mi] * rn1;
    const float u2i = sh2[i0 + mi] * rn2;
    const float a2  = (M == 0) ? -3.f * cc * s * u1i
                    : (M == 1) ? -cc * s * u2i
                               : -3.f * cc * s * u2i;

    // A (16x4 f32), loop-invariant: lanes 0-15 hold K=0,1; lanes 16-31 K=2,3.
    v2f A;
    A.x = hi ? a2  : s * u1i;
    A.y = hi ? 0.f : s * u2i;

    // Diagonal C tile, built once: element (r+8*hi, mi) nonzero iff equal.
    v8f Cdiag;
#pragma unroll
    for (int r = 0; r < 8; ++r)
        Cdiag[r] = ((r + (hi ? 8 : 0)) == mi) ? diag : 0.f;
    const v8f Czero = {};   // hoisted zeros (off-diagonal tiles)

#pragma unroll
    for (int t = 0; t < 8; ++t) {
        const int jt = (wave + t) & 7;   // t==0 -> diagonal tile (jt == wave)
        const int j0 = jt << 4;

        const float u1j = sh1[j0 + mi] * rn1;
        const float u2j = sh2[j0 + mi] * rn2;

        // B rows per M (compile-time picks, no runtime selects on M).
        const float b0lo = (M == 1) ? u1j : u2j;
        const float b1lo = (M == 1) ? u2j : u1j;
        const float b0hi = (M == 2) ? u2j : u1j;

        // B (4x16 f32): lanes 0-15 rows K=0,1; lanes 16-31 rows K=2,3.
        v2f Bm;
        Bm.x = hi ? b0hi : b0lo;
        Bm.y = hi ? 0.f  : b1lo;

        // D = A x B + C : finished 16x16 tile of the scaled Hessian block.
        v8f Dt = __builtin_amdgcn_wmma_f32_16x16x4_f32(
            false, A, false, Bm, (short)0,
            (t == 0) ? Cdiag : Czero, false, false);

        // 8 streaming b32 stores, 512B row stride (write-once output).
#pragma unroll
        for (int r = 0; r < 8; ++r)
            __builtin_nontemporal_store(Dt[r], base + (size_t)r * DIM + j0);
    }
}

// Grid: (B, 3). blockIdx.x = batch row b, blockIdx.y = Hessian block m.
// Block: 256 threads = 8 wave32, one 16-row tile band per wave.
__global__ __launch_bounds__(256) void arccos_hess_wmma(
    const float* __restrict__ z1, const float* __restrict__ z2,
    float* __restrict__ out, int B)
{
    __shared__ float sh1[DIM];
    __shared__ float sh2[DIM];

    const int b    = blockIdx.x;
    const int m    = blockIdx.y;      // scalar -> uniform branch below
    const int tid  = threadIdx.x;
    const int lane = tid & 31;
    const int wave = tid >> 5;
    const int mi   = lane & 15;
    const bool hi  = (lane >= 16);

    // Stage raw vectors into LDS (1 KB).
    if (tid < DIM) {
        sh1[tid] = z1[(size_t)b * DIM + tid];
        sh2[tid] = z2[(size_t)b * DIM + tid];
    }
    __syncthreads();

    // Per-wave redundant reduction over 128 elements (wave32 butterfly).
    float s11 = 0.f, s22 = 0.f, s12 = 0.f;
#pragma unroll
    for (int k = 0; k < 4; ++k) {
        float a = sh1[lane + 32 * k];
        float d = sh2[lane + 32 * k];
        s11 += a * a;
        s22 += d * d;
        s12 += a * d;
    }
#pragma unroll
    for (int x = 16; x >= 1; x >>= 1) {
        s11 += __shfl_xor(s11, x, 32);
        s22 += __shfl_xor(s22, x, 32);
        s12 += __shfl_xor(s12, x, 32);
    }
    const float rn1 = rsqrtf(s11);          // 1/||z1||
    const float rn2 = rsqrtf(s22);          // 1/||z2||
    const float cc  = s12 * rn1 * rn2;      // cos

    // Output base for this (m, b, wave, lane).
    float* const base = out + (size_t)m * ((size_t)B * DIM * DIM)
                            + (size_t)b * (DIM * DIM)
                            + (size_t)((wave << 4) + (hi ? 8 : 0)) * DIM + mi;

    // Uniform scalar branch on m (SCC branch, EXEC untouched).
    if (m == 0)      hess_tiles<0>(sh1, sh2, rn1, rn2, cc, base, wave, mi, hi);
    else if (m == 1) hess_tiles<1>(sh1, sh2, rn1, rn2, cc, base, wave, mi, hi);
    else             hess_tiles<2>(sh1, sh2, rn1, rn2, cc, base, wave, mi, hi);
}

extern "C" void kernel_launch(void* const* d_in, const int* in_sizes, int n_in,
                              void* d_out, int out_size, void* d_ws, size_t ws_size,
                              hipStream_t stream) {
    const float* z1 = (const float*)d_in[0];
    const float* z2 = (const float*)d_in[1];
    float* out = (float*)d_out;
    const int B = in_sizes[0] / DIM;   // 4096
    if (B <= 0) return;
    arccos_hess_wmma<<<dim3(B, 3), dim3(256), 0, stream>>>(z1, z2, out, B);
}